// SlowTrustMemory_72456098283714
// MI455X (gfx1250) — compile-verified
//
#include <hip/hip_runtime.h>
#include <hip/hip_bf16.h>
#include <cmath>

#define B_ 8
#define T_ 4096
#define D_ 1024
#define EPS 0.01f
#define TCHUNK 128
#define NCHUNK (T_ / TCHUNK)   // 32
#define LOG2_099 (-0.014500139f)  // log2(0.99)

typedef __attribute__((ext_vector_type(2))) float v2f;
typedef __attribute__((ext_vector_type(8))) float v8f;

// ---------------------------------------------------------------------------
// K1: per-batch suffix-count of gates -> c[b,t] = eps*g_t*0.99^(#gates after t)
//     also zero final-memory region (needed by atomic fallback path).
// 8 blocks x 128 threads, 32 elements/thread.
// ---------------------------------------------------------------------------
__global__ void k1_coeff(const int* __restrict__ pad, const int* __restrict__ upd,
                         float* __restrict__ cbuf, float* __restrict__ mem_out) {
    const int b   = blockIdx.x;
    const int tid = threadIdx.x;            // 128 threads
    __shared__ int cnt_s[128];
    const int t0 = tid * 32;

    unsigned gbits = 0u;
    int cnt = 0;
#pragma unroll
    for (int i = 0; i < 32; ++i) {
        int t = t0 + i;
        int g = (pad[b * T_ + t] == 0) && (upd[b * T_ + t] != 0);
        gbits |= (unsigned)g << i;
        cnt += g;
    }
    cnt_s[tid] = cnt;
    __syncthreads();
    // inclusive suffix-sum across threads
    for (int off = 1; off < 128; off <<= 1) {
        int add = (tid + off < 128) ? cnt_s[tid + off] : 0;
        int cur = cnt_s[tid];
        __syncthreads();
        cnt_s[tid] = cur + add;
        __syncthreads();
    }
    int S = (tid < 127) ? cnt_s[tid + 1] : 0;   // gates strictly after this chunk

    float cvals[32];
#pragma unroll
    for (int i = 31; i >= 0; --i) {
        int g = (gbits >> i) & 1;
        cvals[i] = g ? (EPS * exp2f(LOG2_099 * (float)S)) : 0.0f;
        S += g;
    }
#pragma unroll
    for (int i = 0; i < 32; ++i) cbuf[b * T_ + t0 + i] = cvals[i];

    for (int i = tid; i < D_; i += 128) mem_out[b * D_ + i] = 0.0f;
}

// ---------------------------------------------------------------------------
// K2: single streaming pass over X (134 MB). Per block: batch b, 128 timesteps,
// full D=1024. For each 16x1024 tile staged in LDS:
//   - per-wave WMMA f32 16x16x4: dot of 16 rows against w over the wave's
//     128-column slice (B fragment = w broadcast across all 16 N columns)
//   - VALU FMA accumulation of c[t]*x[t,d] for the final-memory matvec
// Cross-wave dot reduction via ds_add_f32 into LDS; block owns its (b,t-chunk)
// slice of dbuf so the global write is non-atomic.
// ---------------------------------------------------------------------------
__global__ void __launch_bounds__(256)
k2_stream(const float* __restrict__ x, const float* __restrict__ wvec,
          const float* __restrict__ cbuf, float* __restrict__ dbuf,
          float* __restrict__ mempart, float* __restrict__ mem_out,
          int atomic_mode) {
    __shared__ float tile[16 * 1028];   // stride 1028 -> conflict-free A-frag reads
    __shared__ float wlds[D_];
    __shared__ float clds[TCHUNK];
    __shared__ float dotl[TCHUNK];

    const int tid   = threadIdx.x;
    const int b     = blockIdx.y;
    const int chunk = blockIdx.x;
    const int t0    = chunk * TCHUNK;

    for (int i = tid; i < D_; i += 256) wlds[i] = wvec[i];
    if (tid < TCHUNK) {
        clds[tid] = cbuf[b * T_ + t0 + tid];
        dotl[tid] = 0.0f;
    }

    const int wid     = tid >> 5;
    const int lane    = tid & 31;
    const int row     = lane & 15;           // A-matrix M = row
    const int khalf   = (lane >> 4) << 1;    // lanes 16-31 hold K+2,K+3
    const int colBase = wid * 128 + lane * 4;

    float4 macc = make_float4(0.f, 0.f, 0.f, 0.f);

    for (int tt = 0; tt < 8; ++tt) {
        __syncthreads();
        // cooperative coalesced load of 16x1024 tile (float4)
        const float4* src =
            (const float4*)(x + ((size_t)b * T_ + (size_t)(t0 + tt * 16)) * D_);
        for (int i = tid; i < 16 * 256; i += 256) {
            int r = i >> 8, c4 = i & 255;
            *(float4*)(tile + r * 1028 + (c4 << 2)) = src[(r << 8) + c4];
        }
        __syncthreads();

        // --- WMMA dot: acc[t][n] += sum_k X[t,k]*w[k] (all n columns equal) ---
        v8f acc = {};
        const int kend = wid * 128 + 128;
        for (int k0 = wid * 128; k0 < kend; k0 += 4) {
            const int kk = k0 + khalf;
            v2f av = *(const v2f*)(tile + row * 1028 + kk);
            v2f bv;
            bv[0] = wlds[kk];        // B[k][n] = w[k] broadcast over n
            bv[1] = wlds[kk + 1];
            acc = __builtin_amdgcn_wmma_f32_16x16x4_f32(
                false, av, false, bv, (short)0, acc, false, false);
        }
        // C layout: lanes 0-15 vgpr r -> M=r ; lanes 16-31 vgpr r -> M=8+r
        if ((lane & 15) == 0) {
            const int half8 = (lane >> 4) << 3;
#pragma unroll
            for (int r = 0; r < 8; ++r)
                atomicAdd(&dotl[tt * 16 + half8 + r], acc[r]);   // ds_add_f32
        }

        // --- final-memory accumulation: macc += c[t] * X[t, colBase..+3] ---
#pragma unroll
        for (int r = 0; r < 16; ++r) {
            float cc = clds[tt * 16 + r];          // wave-uniform
            if (cc != 0.0f) {
                const float4 xv = *(const float4*)(tile + r * 1028 + colBase);
                macc.x += cc * xv.x;
                macc.y += cc * xv.y;
                macc.z += cc * xv.z;
                macc.w += cc * xv.w;
            }
        }
    }
    __syncthreads();

    if (tid < TCHUNK) dbuf[b * T_ + t0 + tid] = dotl[tid];

    if (atomic_mode) {
        atomicAdd(&mem_out[b * D_ + colBase + 0], macc.x);
        atomicAdd(&mem_out[b * D_ + colBase + 1], macc.y);
        atomicAdd(&mem_out[b * D_ + colBase + 2], macc.z);
        atomicAdd(&mem_out[b * D_ + colBase + 3], macc.w);
    } else {
        float* p = mempart + (size_t)(b * NCHUNK + chunk) * D_ + colBase;
        p[0] = macc.x; p[1] = macc.y; p[2] = macc.z; p[3] = macc.w;
    }
}

// ---------------------------------------------------------------------------
// K4: deterministic reduction of per-chunk memory partials.
// ---------------------------------------------------------------------------
__global__ void k4_reduce(const float* __restrict__ mempart,
                          float* __restrict__ mem_out) {
    int idx = blockIdx.x * blockDim.x + threadIdx.x;  // over B_*D_
    int b = idx >> 10, d = idx & 1023;
    float s = 0.f;
#pragma unroll 4
    for (int ch = 0; ch < NCHUNK; ++ch)
        s += mempart[((size_t)(b * NCHUNK + ch) << 10) + d];
    mem_out[idx] = s;
}

// ---------------------------------------------------------------------------
// K3: per-batch parallel affine scan  s_t = a_t*s_{t-1} + b_t,
//     a_t = 1-eps*g, b_t = eps*g*d_t ; then bias = clamp(pad?1:1+tanh(s)).
// 8 blocks x 128 threads, 32 elements/thread, Hillis-Steele over chunks.
// ---------------------------------------------------------------------------
__global__ void k3_scan(const float* __restrict__ dbuf, const int* __restrict__ pad,
                        const int* __restrict__ upd, float* __restrict__ bias_out) {
    const int b   = blockIdx.x;
    const int tid = threadIdx.x;   // 128
    __shared__ float As[128], Bs[128];
    const int t0 = tid * 32;

    unsigned gbits = 0u, pbits = 0u;
    float bloc[32];
    float A = 1.0f, Bc = 0.0f;
#pragma unroll
    for (int i = 0; i < 32; ++i) {
        int t = t0 + i;
        int p = (pad[b * T_ + t] != 0);
        int g = (!p) && (upd[b * T_ + t] != 0);
        pbits |= (unsigned)p << i;
        gbits |= (unsigned)g << i;
        float a  = g ? (1.0f - EPS) : 1.0f;
        float bb = g ? (EPS * dbuf[b * T_ + t]) : 0.0f;
        bloc[i] = bb;
        A  = a * A;
        Bc = a * Bc + bb;
    }
    As[tid] = A; Bs[tid] = Bc;
    __syncthreads();
    for (int off = 1; off < 128; off <<= 1) {
        float pA = 1.0f, pB = 0.0f;
        if (tid >= off) { pA = As[tid - off]; pB = Bs[tid - off]; }
        float cA = As[tid], cB = Bs[tid];
        __syncthreads();
        As[tid] = cA * pA;
        Bs[tid] = cA * pB + cB;
        __syncthreads();
    }
    float s = (tid > 0) ? Bs[tid - 1] : 0.0f;   // initial memory is zero
#pragma unroll
    for (int i = 0; i < 32; ++i) {
        float a = ((gbits >> i) & 1) ? (1.0f - EPS) : 1.0f;
        s = a * s + bloc[i];
        float bias = 1.0f + tanhf(s);
        if ((pbits >> i) & 1) bias = 1.0f;
        bias = fminf(fmaxf(bias, 0.8f), 1.2f);
        bias_out[b * T_ + t0 + i] = bias;
    }
}

// ---------------------------------------------------------------------------
extern "C" void kernel_launch(void* const* d_in, const int* in_sizes, int n_in,
                              void* d_out, int out_size, void* d_ws, size_t ws_size,
                              hipStream_t stream) {
    const float* x  = (const float*)d_in[0];   // write_signal (B,T,D) f32
    const int*  pad = (const int*)d_in[1];     // pad_mask (B,T)
    const int*  upd = (const int*)d_in[2];     // update_mask (B,T)
    const float* w  = (const float*)d_in[3];   // w (D,)

    float* bias_out = (float*)d_out;                        // (B,T)
    float* mem_out  = (float*)d_out + (size_t)B_ * T_;      // (B,D)

    float* dbuf    = (float*)d_ws;                          // d[b,t] = x_t.w
    float* cbuf    = dbuf + (size_t)B_ * T_;                // c[b,t]
    float* mempart = cbuf + (size_t)B_ * T_;                // per-chunk partials

    const size_t need =
        ((size_t)B_ * T_ * 2 + (size_t)B_ * NCHUNK * D_) * sizeof(float);
    const int atomic_mode = (ws_size < need) ? 1 : 0;

    k1_coeff<<<dim3(B_), 128, 0, stream>>>(pad, upd, cbuf, mem_out);
    k2_stream<<<dim3(NCHUNK, B_), 256, 0, stream>>>(x, w, cbuf, dbuf, mempart,
                                                    mem_out, atomic_mode);
    if (!atomic_mode)
        k4_reduce<<<dim3((B_ * D_) / 256), 256, 0, stream>>>(mempart, mem_out);
    k3_scan<<<dim3(B_), 128, 0, stream>>>(dbuf, pad, upd, bias_out);
}